// MLPScorer_7447473291455
// MI455X (gfx1250) — compile-verified
//
#include <hip/hip_runtime.h>
#include <hip/hip_bf16.h>

typedef float v2f __attribute__((ext_vector_type(2)));
typedef float v8f __attribute__((ext_vector_type(8)));

#define NN 512
#define DD 256
#define HH 512

// fast, saturating tanh: 1 - 2/(1 + exp2(x * 2*log2(e)))
__device__ __forceinline__ float tanh_fast(float x) {
    float e = __builtin_amdgcn_exp2f(x * 2.885390081777927f); // 2/ln(2)
    float r = __builtin_amdgcn_rcpf(1.0f + e);
    return __builtin_fmaf(-2.0f, r, 1.0f);
}

// Kernel 1: [xa' | xb] = x @ Wc  (M=512, N=1024, K=256) with V_WMMA_F32_16X16X4_F32.
//   Wc[d, n] = (n < 512) ? W1[n, d] : W1[n-512, 256+d]
//   xa'[i,h] = xa[i,h] + b1[h]  (bias folded here), xb stored raw.
// One wave -> one 16x16 output tile. 8 waves/block. grid = (8, 32): 64 N-tiles x 32 M-tiles.
__global__ __launch_bounds__(256) void mlpscore_gemm_wmma(
    const float* __restrict__ x, const float* __restrict__ W1,
    const float* __restrict__ b1, float* __restrict__ xa_out,
    float* __restrict__ xb_out) {
    const int lane = threadIdx.x & 31;
    const int wave = threadIdx.x >> 5;
    const int n0 = (blockIdx.x * 8 + wave) * 16;  // 0..1008
    const int m0 = blockIdx.y * 16;               // 0..496

    const int l15 = lane & 15;
    const int kb  = (lane >> 4) << 1;  // 0 or 2 (K sub-offset per CDNA5 ISA layout)

    // A fragment source: x[m0 + l15][k + kb .. +1]
    const float* a_ptr = x + (m0 + l15) * DD + kb;
    // B fragment source: Wc[k + kb .. +1][n0 + l15] -> contiguous within a W1 row
    const int nc = n0 + l15;
    const float* b_ptr = (nc < HH) ? (W1 + nc * (2 * DD) + kb)
                                   : (W1 + (nc - HH) * (2 * DD) + DD + kb);

    v8f c = {};
#pragma unroll 8
    for (int k = 0; k < DD; k += 4) {
        v2f a = *(const v2f*)(a_ptr + k);
        v2f b = *(const v2f*)(b_ptr + k);
        c = __builtin_amdgcn_wmma_f32_16x16x4_f32(
            /*neg_a=*/false, a, /*neg_b=*/false, b,
            /*c_mod=*/(short)0, c, /*reuse_a=*/false, /*reuse_b=*/false);
    }

    // D layout: VGPR r -> row m0 + r + 8*(lane>=16), col n0 + (lane&15)
    const int mbase = m0 + ((lane >> 4) << 3);
    if (nc < HH) {
        const float bias = b1[nc];
#pragma unroll
        for (int r = 0; r < 8; ++r)
            xa_out[(mbase + r) * HH + nc] = c[r] + bias;
    } else {
        const int nb = nc - HH;
#pragma unroll
        for (int r = 0; r < 8; ++r)
            xb_out[(mbase + r) * HH + nb] = c[r];
    }
}

// Kernel 2: score[i,j] = sum_h tanh(xa'[i,h] + xb[j,h]) * W2[h] + b2
// 16x16 (i,j) tile per 256-thread block; h chunked by 128 through LDS.
__global__ __launch_bounds__(256) void mlpscore_tanh_reduce(
    const float* __restrict__ xa, const float* __restrict__ xb,
    const float* __restrict__ W2, const float* __restrict__ b2,
    float* __restrict__ out) {
    __shared__ float sa[16][128];
    __shared__ float sb[16][128];
    __shared__ float sw[128];

    const int i0 = blockIdx.y * 16;
    const int j0 = blockIdx.x * 16;
    const int ti = threadIdx.x >> 4;  // i within tile
    const int tj = threadIdx.x & 15;  // j within tile

    float acc = 0.0f;
    for (int h0 = 0; h0 < HH; h0 += 128) {
        __syncthreads();  // previous chunk fully consumed
        // cooperative float4 loads: 512 float4 per array, 2 per thread each
        for (int t = threadIdx.x; t < 512; t += 256) {
            const int r = t >> 5, c4 = (t & 31) << 2;
            *(float4*)(&sa[r][c4]) = *(const float4*)(xa + (i0 + r) * HH + h0 + c4);
            *(float4*)(&sb[r][c4]) = *(const float4*)(xb + (j0 + r) * HH + h0 + c4);
        }
        if (threadIdx.x < 32)
            *(float4*)(&sw[threadIdx.x << 2]) =
                *(const float4*)(W2 + h0 + (threadIdx.x << 2));
        __syncthreads();

#pragma unroll 4
        for (int c = 0; c < 128; ++c) {
            float v = sa[ti][c] + sb[tj][c];
            acc = __builtin_fmaf(tanh_fast(v), sw[c], acc);
        }
    }
    out[(i0 + ti) * NN + (j0 + tj)] = acc + b2[0];
}

extern "C" void kernel_launch(void* const* d_in, const int* in_sizes, int n_in,
                              void* d_out, int out_size, void* d_ws, size_t ws_size,
                              hipStream_t stream) {
    const float* x  = (const float*)d_in[0];  // (512,1,256) f32
    const float* W1 = (const float*)d_in[1];  // (512,512) f32
    const float* b1 = (const float*)d_in[2];  // (512,) f32
    const float* W2 = (const float*)d_in[3];  // (1,512) f32
    const float* b2 = (const float*)d_in[4];  // (1,) f32
    float* out = (float*)d_out;               // (512,512) f32

    float* xa = (float*)d_ws;                 // 512*512 f32 (bias-folded)
    float* xb = xa + NN * HH;                 // 512*512 f32 (needs 2 MB of ws)

    dim3 g1(8, 32);   // 64 N-tiles (8 waves/block) x 32 M-tiles
    mlpscore_gemm_wmma<<<g1, 256, 0, stream>>>(x, W1, b1, xa, xb);

    dim3 g2(32, 32);  // 16x16 (j,i) tiles
    mlpscore_tanh_reduce<<<g2, 256, 0, stream>>>(xa, xb, W2, b2, out);
}